// GNNBackbone_7327214207620
// MI455X (gfx1250) — compile-verified
//
#include <hip/hip_runtime.h>
#include <stdint.h>

#define DFEAT 128

typedef __attribute__((ext_vector_type(2))) float v2f;
typedef __attribute__((ext_vector_type(8))) float v8f;

// ---------------- CSR build ----------------

__global__ __launch_bounds__(256) void k_init(int* __restrict__ indeg, int* __restrict__ cursor, int n) {
    int i = blockIdx.x * blockDim.x + threadIdx.x;
    if (i < n) { indeg[i] = 0; cursor[i] = 0; }
}

__global__ __launch_bounds__(256) void k_count(const int* __restrict__ dst, int* __restrict__ indeg, int ne) {
    int e = blockIdx.x * blockDim.x + threadIdx.x;
    if (e < ne) atomicAdd(&indeg[dst[e]], 1);
}

__global__ __launch_bounds__(256) void k_dis(const int* __restrict__ indeg, float* __restrict__ dis, int n) {
    int i = blockIdx.x * blockDim.x + threadIdx.x;
    if (i < n) dis[i] = rsqrtf((float)(indeg[i] + 1));   // +1 self-loop; always >= 1
}

// single-block chunked inclusive scan: row_ptr[i+1] = sum_{j<=i} indeg[j], row_ptr[0] = 0
__global__ __launch_bounds__(1024) void k_scan(const int* __restrict__ indeg, int* __restrict__ row_ptr, int n) {
    __shared__ int sh[1024];
    __shared__ int carry;
    const int t = threadIdx.x;
    if (t == 0) { carry = 0; row_ptr[0] = 0; }
    __syncthreads();
    for (int base = 0; base < n; base += 1024) {
        const int idx = base + t;
        sh[t] = (idx < n) ? indeg[idx] : 0;
        __syncthreads();
        for (int off = 1; off < 1024; off <<= 1) {
            int xv = (t >= off) ? sh[t - off] : 0;
            __syncthreads();
            sh[t] += xv;
            __syncthreads();
        }
        const int inc = sh[t] + carry;
        if (idx < n) row_ptr[idx + 1] = inc;
        __syncthreads();
        if (t == 1023) carry = inc;
        __syncthreads();
    }
}

__global__ __launch_bounds__(256) void k_fill(const int* __restrict__ src, const int* __restrict__ dst,
                                              const int* __restrict__ row_ptr, int* __restrict__ cursor,
                                              int* __restrict__ csr_src, int ne) {
    int e = blockIdx.x * blockDim.x + threadIdx.x;
    if (e < ne) {
        int d = dst[e];
        int pos = row_ptr[d] + atomicAdd(&cursor[d], 1);
        csr_src[pos] = src[e];
    }
}

// stable per-row order (insertion sort, avg degree 16) -> deterministic fp summation order
__global__ __launch_bounds__(256) void k_sortrows(int* __restrict__ csr_src, const int* __restrict__ row_ptr, int n) {
    int i = blockIdx.x * blockDim.x + threadIdx.x;
    if (i >= n) return;
    int beg = row_ptr[i], end = row_ptr[i + 1];
    for (int a = beg + 1; a < end; ++a) {
        int v = csr_src[a];
        int b = a - 1;
        while (b >= beg && csr_src[b] > v) { csr_src[b + 1] = csr_src[b]; --b; }
        csr_src[b + 1] = v;
    }
}

// ---------------- dense GEMM: out[n x 128] = H[n x 128] @ W[128 x 128], fp32 WMMA ----------------

__global__ __launch_bounds__(256) void k_gemm(const float* __restrict__ H, const float* __restrict__ W,
                                              float* __restrict__ out, int n) {
    __shared__ float As[16][132];            // padded: conflict-free, 8B-aligned rows
    const int m0 = blockIdx.x * 16;
    const int t  = threadIdx.x;
    {   // cooperative 16x128 A-tile load: thread t -> row t/16, 8 consecutive floats
        int r = t >> 4;
        int c = (t & 15) * 8;
        int row = m0 + r; if (row >= n) row = n - 1;          // clamp (n % 16 == 0 here)
        const float* sp = H + (size_t)row * DFEAT + c;
        #pragma unroll
        for (int q = 0; q < 8; ++q) As[r][c + q] = sp[q];
    }
    __syncthreads();

    const int lane = t & 31;
    const int ln   = lane & 15;       // M (for A) / N (for B,C)
    const int half = lane >> 4;       // lane half selects K pair / M+8
    const int n0   = (t >> 5) * 16;   // each of 8 waves owns a 16-col tile -> full 128 cols

    v8f acc = {0.f, 0.f, 0.f, 0.f, 0.f, 0.f, 0.f, 0.f};
    #pragma unroll
    for (int kk = 0; kk < DFEAT; kk += 4) {
        const int ka = kk + half * 2;                 // lanes 0-15: K={kk,kk+1}; 16-31: K={kk+2,kk+3}
        v2f a = { As[ln][ka], As[ln][ka + 1] };
        v2f b = { W[(size_t)ka * DFEAT + n0 + ln],
                  W[(size_t)(ka + 1) * DFEAT + n0 + ln] };
        acc = __builtin_amdgcn_wmma_f32_16x16x4_f32(false, a, false, b, (short)0, acc, false, false);
    }
    #pragma unroll
    for (int r = 0; r < 8; ++r) {                     // VGPR r holds M = r + 8*half
        int row = m0 + half * 8 + r;
        if (row < n) out[(size_t)row * DFEAT + n0 + ln] = acc[r];
    }
}

// ---------------- CSR gather aggregation + bias (+ReLU): one wave per node, float4 per lane ----------------

__global__ __launch_bounds__(128) void k_agg(const float* __restrict__ hW, const float* __restrict__ dis,
                                             const int* __restrict__ row_ptr, const int* __restrict__ csr_src,
                                             const float* __restrict__ bias, float* __restrict__ out,
                                             int n, int relu) {
    const int lane = threadIdx.x & 31;
    const int i    = blockIdx.x * 4 + (threadIdx.x >> 5);
    if (i >= n) return;
    const float4* hw4 = (const float4*)hW;
    const float di = dis[i];
    const float ws = di * di;                          // self-loop norm = 1/deg
    float4 v = hw4[(size_t)i * 32 + lane];
    float ax = ws * v.x, ay = ws * v.y, az = ws * v.z, aw = ws * v.w;
    const int beg = row_ptr[i], end = row_ptr[i + 1];
    for (int e = beg; e < end; ++e) {
        int s = csr_src[e];                            // wave-uniform -> scalar load path
        float ns = di * dis[s];
        float4 u = hw4[(size_t)s * 32 + lane];
        ax = fmaf(ns, u.x, ax); ay = fmaf(ns, u.y, ay);
        az = fmaf(ns, u.z, az); aw = fmaf(ns, u.w, aw);
    }
    const float4 bb = ((const float4*)bias)[lane];
    ax += bb.x; ay += bb.y; az += bb.z; aw += bb.w;
    if (relu) { ax = fmaxf(ax, 0.f); ay = fmaxf(ay, 0.f); az = fmaxf(az, 0.f); aw = fmaxf(aw, 0.f); }
    float4 r; r.x = ax; r.y = ay; r.z = az; r.w = aw;
    ((float4*)out)[(size_t)i * 32 + lane] = r;
}

// ---------------- host glue ----------------

extern "C" void kernel_launch(void* const* d_in, const int* in_sizes, int n_in,
                              void* d_out, int out_size, void* d_ws, size_t ws_size,
                              hipStream_t stream) {
    const float* x  = (const float*)d_in[0];
    const int*   ei = (const int*)d_in[1];
    const float* Wl[3] = {(const float*)d_in[2], (const float*)d_in[4], (const float*)d_in[6]};
    const float* bl[3] = {(const float*)d_in[3], (const float*)d_in[5], (const float*)d_in[7]};
    const int n  = in_sizes[0] / DFEAT;     // 50000
    const int ne = in_sizes[1] / 2;         // 800000
    const int* src = ei;
    const int* dst = ei + ne;
    float* out = (float*)d_out;

    uintptr_t p = (uintptr_t)d_ws;
    auto alloc = [&](size_t bytes) -> void* {
        uintptr_t q = (p + 15) & ~(uintptr_t)15;
        p = q + bytes;
        return (void*)q;
    };
    int*   indeg   = (int*)  alloc((size_t)n * 4);
    int*   cursor  = (int*)  alloc((size_t)n * 4);
    int*   row_ptr = (int*)  alloc((size_t)(n + 1) * 4);
    int*   csr_src = (int*)  alloc((size_t)ne * 4);
    float* dis     = (float*)alloc((size_t)n * 4);
    float* bufA    = (float*)alloc((size_t)n * DFEAT * 4);   // total ws ~ 30 MB

    const dim3 b256(256);
    k_init    <<<dim3((n  + 255) / 256), b256, 0, stream>>>(indeg, cursor, n);
    k_count   <<<dim3((ne + 255) / 256), b256, 0, stream>>>(dst, indeg, ne);
    k_dis     <<<dim3((n  + 255) / 256), b256, 0, stream>>>(indeg, dis, n);
    k_scan    <<<dim3(1), dim3(1024),    0, stream>>>(indeg, row_ptr, n);
    k_fill    <<<dim3((ne + 255) / 256), b256, 0, stream>>>(src, dst, row_ptr, cursor, csr_src, ne);
    k_sortrows<<<dim3((n  + 255) / 256), b256, 0, stream>>>(csr_src, row_ptr, n);

    const dim3 gGemm((n + 15) / 16);
    const dim3 gAgg((n + 3) / 4);
    const dim3 bAgg(128);
    // layer 0: x -> bufA -> out (ReLU)
    k_gemm<<<gGemm, b256, 0, stream>>>(x,   Wl[0], bufA, n);
    k_agg <<<gAgg, bAgg, 0, stream>>>(bufA, dis, row_ptr, csr_src, bl[0], out, n, 1);
    // layer 1: out -> bufA -> out (ReLU)
    k_gemm<<<gGemm, b256, 0, stream>>>(out, Wl[1], bufA, n);
    k_agg <<<gAgg, bAgg, 0, stream>>>(bufA, dis, row_ptr, csr_src, bl[1], out, n, 1);
    // layer 2: out -> bufA -> out (no ReLU)
    k_gemm<<<gGemm, b256, 0, stream>>>(out, Wl[2], bufA, n);
    k_agg <<<gAgg, bAgg, 0, stream>>>(bufA, dis, row_ptr, csr_src, bl[2], out, n, 0);
}